// feature_embedding_27702539059310
// MI455X (gfx1250) — compile-verified
//
#include <hip/hip_runtime.h>

#define B_ROWS 16384
#define N_COLS 32          // NUM_COLS + CAT_COLS
#define N_NUM  16
#define HID    128
#define FEAT   4096        // 32 * 128
#define VOCAB  1000
#define NCHUNK 32
#define ROWS_PER_CHUNK (B_ROWS / NCHUNK)   // 512

typedef float v2f  __attribute__((ext_vector_type(2)));
typedef float v8f  __attribute__((ext_vector_type(8)));
typedef float f32x4 __attribute__((ext_vector_type(4)));

// ---------------------------------------------------------------------------
// Pass 1: per-chunk partial sum / sum-of-squares per feature, batch reduction
// on the matrix pipe via V_WMMA_F32_16X16X4_F32 with A == ones:
//   D[m,n] = sum_k B[k,n] + C[m,n]   (every row of D carries the column sums)
// B tile layout (16x16x4, 32-bit B): VGPR0 lanes0-15 = K0, lanes16-31 = K2;
//                                    VGPR1 lanes0-15 = K1, lanes16-31 = K3.
// Wave owns 16 features (N = lane&15), 4 batch rows per WMMA pair.
// num/cat split is wave-uniform; readfirstlane forces a scalar branch so EXEC
// stays all-ones through the WMMA loops (ISA requirement, no saveexec churn).
// Gathers use 32-bit offsets (idx*512B <= 512KB) so the compiler emits the
// SGPR-base + VGPR-offset (scale_offset) addressing mode, no 64-bit VALU adds.
// ---------------------------------------------------------------------------
__global__ __launch_bounds__(128)
void fe_stats_kernel(const float* __restrict__ in, const float* __restrict__ W,
                     const float* __restrict__ bias, const float* __restrict__ T,
                     float* __restrict__ psum, float* __restrict__ psumsq)
{
    const int wave  = threadIdx.x >> 5;
    const int lane  = threadIdx.x & 31;
    const int n     = lane & 15;
    const int khalf = lane >> 4;                 // 0: rows k,k+1  1: rows k+2,k+3
    const int fbase = blockIdx.x * 64 + wave * 16;
    const int f     = fbase + n;
    const int c     = f >> 7;                    // column, uniform per wave
    const int h     = f & 127;

    const int row0 = blockIdx.y * ROWS_PER_CHUNK;
    v2f ones; ones[0] = 1.0f; ones[1] = 1.0f;
    v8f accS = {};
    v8f accQ = {};

    if (__builtin_amdgcn_readfirstlane(c < N_NUM ? 1 : 0)) {
        const float w  = W[c * HID + h];
        const float bb = bias[c * HID + h];
#pragma unroll 2
        for (int k = 0; k < ROWS_PER_CHUNK; k += 4) {
            const int b0 = row0 + k + khalf * 2;
            const float x0 = in[b0 * N_COLS + c];
            const float x1 = in[(b0 + 1) * N_COLS + c];
            const float v0 = fmaxf(fmaf(x0, w, bb), 0.f);
            const float v1 = fmaxf(fmaf(x1, w, bb), 0.f);
            v2f bm; bm[0] = v0;      bm[1] = v1;
            v2f bq; bq[0] = v0 * v0; bq[1] = v1 * v1;
            accS = __builtin_amdgcn_wmma_f32_16x16x4_f32(false, ones, false, bm,
                                                         (short)0, accS, false, false);
            accQ = __builtin_amdgcn_wmma_f32_16x16x4_f32(false, ones, false, bq,
                                                         (short)0, accQ, false, false);
        }
    } else {
        // Per-column table base stays scalar; per-row offset is a 32-bit int.
        const float* tbl = T + (size_t)(c - N_NUM) * (VOCAB * HID) + h;
#pragma unroll 2
        for (int k = 0; k < ROWS_PER_CHUNK; k += 4) {
            const int b0 = row0 + k + khalf * 2;
            const int i0 = (int)in[b0 * N_COLS + c];
            const int i1 = (int)in[(b0 + 1) * N_COLS + c];
            const float v0 = tbl[i0 * HID];
            const float v1 = tbl[i1 * HID];
            v2f bm; bm[0] = v0;      bm[1] = v1;
            v2f bq; bq[0] = v0 * v0; bq[1] = v1 * v1;
            accS = __builtin_amdgcn_wmma_f32_16x16x4_f32(false, ones, false, bm,
                                                         (short)0, accS, false, false);
            accQ = __builtin_amdgcn_wmma_f32_16x16x4_f32(false, ones, false, bq,
                                                         (short)0, accQ, false, false);
        }
    }

    // D row M=0 (VGPR0, lanes 0-15) holds the per-feature column sums.
    if (lane < 16) {
        psum  [blockIdx.y * FEAT + f] = accS[0];
        psumsq[blockIdx.y * FEAT + f] = accQ[0];
    }
}

// ---------------------------------------------------------------------------
// Pass 2: fold chunk partials + gamma/beta into per-feature scale/shift.
// ---------------------------------------------------------------------------
__global__ __launch_bounds__(128)
void fe_finalize_kernel(const float* __restrict__ psum, const float* __restrict__ psumsq,
                        const float* __restrict__ gamma, const float* __restrict__ beta,
                        float* __restrict__ scale, float* __restrict__ shift)
{
    const int f = blockIdx.x * 128 + threadIdx.x;
    float s = 0.f, q = 0.f;
#pragma unroll
    for (int ch = 0; ch < NCHUNK; ++ch) {
        s += psum  [ch * FEAT + f];
        q += psumsq[ch * FEAT + f];
    }
    const float invB = 1.0f / (float)B_ROWS;
    const float mean = s * invB;
    const float var  = fmaf(q, invB, -mean * mean);
    const float inv  = rsqrtf(var + 1e-5f);
    const float sc   = gamma[f] * inv;
    scale[f] = sc;
    shift[f] = fmaf(-mean, sc, beta[f]);
}

// ---------------------------------------------------------------------------
// Pass 3: recompute fe fused with normalization; 16B gathers and non-temporal
// 16B stores so the 256 MB write stream doesn't evict L2-resident tables.
// num/cat split is wave-uniform -> scalar branch hoisted out of the row loop;
// gathers use 32-bit offsets for GVS addressing.
// ---------------------------------------------------------------------------
__global__ __launch_bounds__(128)
void fe_out_kernel(const float* __restrict__ in, const float* __restrict__ W,
                   const float* __restrict__ bias, const float* __restrict__ T,
                   const float* __restrict__ scale, const float* __restrict__ shift,
                   float* __restrict__ out)
{
    const int f4 = blockIdx.x * 128 + threadIdx.x;   // float4 index, 0..1023
    const int f  = f4 * 4;
    const int c  = f >> 7;                           // uniform per wave
    const int h  = f & 127;

    const f32x4 sc4 = *(const f32x4*)(scale + f);
    const f32x4 sh4 = *(const f32x4*)(shift + f);

    const int rowsPerBlock = B_ROWS / gridDim.y;
    const int row0 = blockIdx.y * rowsPerBlock;

    if (__builtin_amdgcn_readfirstlane(c < N_NUM ? 1 : 0)) {
        const f32x4 w4 = *(const f32x4*)(W + c * HID + h);
        const f32x4 b4 = *(const f32x4*)(bias + c * HID + h);
#pragma unroll 2
        for (int r = 0; r < rowsPerBlock; ++r) {
            const int b = row0 + r;
            const float x = in[b * N_COLS + c];
            f32x4 v;
            v.x = fmaxf(fmaf(x, w4.x, b4.x), 0.f);
            v.y = fmaxf(fmaf(x, w4.y, b4.y), 0.f);
            v.z = fmaxf(fmaf(x, w4.z, b4.z), 0.f);
            v.w = fmaxf(fmaf(x, w4.w, b4.w), 0.f);
            v.x = fmaf(v.x, sc4.x, sh4.x);
            v.y = fmaf(v.y, sc4.y, sh4.y);
            v.z = fmaf(v.z, sc4.z, sh4.z);
            v.w = fmaf(v.w, sc4.w, sh4.w);
            __builtin_nontemporal_store(v, (f32x4*)(out + (size_t)b * FEAT + f));
        }
    } else {
        const float* tbl = T + (size_t)(c - N_NUM) * (VOCAB * HID) + h;
#pragma unroll 2
        for (int r = 0; r < rowsPerBlock; ++r) {
            const int b = row0 + r;
            const int idx = (int)in[b * N_COLS + c];
            f32x4 v = *(const f32x4*)(tbl + idx * HID);
            v.x = fmaf(v.x, sc4.x, sh4.x);
            v.y = fmaf(v.y, sc4.y, sh4.y);
            v.z = fmaf(v.z, sc4.z, sh4.z);
            v.w = fmaf(v.w, sc4.w, sh4.w);
            __builtin_nontemporal_store(v, (f32x4*)(out + (size_t)b * FEAT + f));
        }
    }
}

extern "C" void kernel_launch(void* const* d_in, const int* in_sizes, int n_in,
                              void* d_out, int out_size, void* d_ws, size_t ws_size,
                              hipStream_t stream) {
    const float* in    = (const float*)d_in[0];   // (16384, 32)
    const float* W     = (const float*)d_in[1];   // (16, 128)
    const float* bias  = (const float*)d_in[2];   // (16, 128)
    const float* T     = (const float*)d_in[3];   // (16, 1000, 128)
    const float* gamma = (const float*)d_in[4];   // (4096,)
    const float* beta  = (const float*)d_in[5];   // (4096,)
    float* out = (float*)d_out;

    float* ws     = (float*)d_ws;
    float* psum   = ws;                           // NCHUNK * FEAT
    float* psumsq = ws + NCHUNK * FEAT;           // NCHUNK * FEAT
    float* scale  = ws + 2 * NCHUNK * FEAT;       // FEAT
    float* shift  = scale + FEAT;                 // FEAT

    // Pass 1: 64 feature-groups x 32 row-chunks, 128 threads (4 wave32).
    fe_stats_kernel<<<dim3(64, NCHUNK), 128, 0, stream>>>(in, W, bias, T, psum, psumsq);
    // Pass 2: 4096 features.
    fe_finalize_kernel<<<dim3(FEAT / 128), 128, 0, stream>>>(psum, psumsq, gamma, beta,
                                                             scale, shift);
    // Pass 3: 8 feature-groups x 256 row-blocks (64 rows each), 128 threads.
    fe_out_kernel<<<dim3(8, 256), 128, 0, stream>>>(in, W, bias, T, scale, shift, out);
}